// P3dPose_66039417143598
// MI455X (gfx1250) — compile-verified
//
#include <hip/hip_runtime.h>

// ---------------------------------------------------------------------------
// Forward kinematics over a 26-bone tree for 131072 poses.
// Bandwidth-bound (~0.8 flop/byte): use the CDNA5 Tensor Data Mover to stream
// each block's contiguous pose window Global->LDS and LDS->Global, compute the
// serial per-pose chain in LDS in place.
// ---------------------------------------------------------------------------

#define NB 26
#define POSES_PER_BLOCK 32
#define CH_F (NB * 9)   // 234 floats per pose (changes / rot matrices)
#define LOC_F (NB * 3)  // 78 floats per pose (locations)

typedef unsigned int u32;
typedef unsigned long long u64;
typedef __attribute__((ext_vector_type(4))) u32 u32x4;
typedef __attribute__((ext_vector_type(8))) int i32x8;
typedef __attribute__((ext_vector_type(4))) int i32x4;

// Build a 1-D TDM descriptor (D# groups 0 and 1) for `ndw` dwords.
// Bit layout per CDNA5 ISA 8.3/8.4:
//  g0: [1:0]=count=1, [63:32]=lds_addr, [120:64]=global_addr, [127:126]=type=2
//  g1: [17:16]=data_size(2 => 4B), [79:48]=tensor_dim0, [127:112]=tile_dim0,
//      tile_dim1..2 = 0 (unused -> 1-D), [207:160]=tensor_dim0_stride
__device__ __forceinline__ void tdm_desc_1d(u32x4 &g0, i32x8 &g1,
                                            u32 lds_off, const void *gptr,
                                            u32 ndw) {
  u64 ga = (u64)gptr;
  g0[0] = 1u;                                            // count=1, user mode
  g0[1] = lds_off;                                       // lds_addr (bytes)
  g0[2] = (u32)ga;                                       // global_addr[31:0]
  g0[3] = ((u32)(ga >> 32) & 0x01FFFFFFu) | 0x80000000u; // ga[56:32] | type=2

  g1[0] = (int)(2u << 16);                  // workgroup_mask=0, data_size=4B
  g1[1] = (int)((ndw & 0xFFFFu) << 16);     // tensor_dim0[15:0]
  g1[2] = (int)((ndw >> 16) & 0xFFFFu);     // tensor_dim0[31:16], tdim1 lo = 0
  g1[3] = (int)((ndw & 0xFFFFu) << 16);     // tile_dim0 (ndw < 65536 here)
  g1[4] = 0;                                // tile_dim1 = tile_dim2 = 0
  g1[5] = (int)ndw;                         // tensor_dim0_stride (unused, 1-D)
  g1[6] = 0;
  g1[7] = 0;
}

__device__ __forceinline__ void tdm_load(u32 lds_off, const void *gptr, u32 ndw) {
  u32x4 g0; i32x8 g1;
  tdm_desc_1d(g0, g1, lds_off, gptr, ndw);
  i32x4 z4 = {0, 0, 0, 0};
  i32x8 z8 = {0, 0, 0, 0, 0, 0, 0, 0};
  __builtin_amdgcn_tensor_load_to_lds(g0, g1, z4, z4, z8, 0);
}

__device__ __forceinline__ void tdm_store(u32 lds_off, void *gptr, u32 ndw) {
  u32x4 g0; i32x8 g1;
  tdm_desc_1d(g0, g1, lds_off, gptr, ndw);
  i32x4 z4 = {0, 0, 0, 0};
  i32x8 z8 = {0, 0, 0, 0, 0, 0, 0, 0};
  __builtin_amdgcn_tensor_store_from_lds(g0, g1, z4, z4, z8, 0);
}

// Low 32 bits of a generic LDS pointer are the workgroup-relative byte offset
// (CDNA5 ISA 10.2: LDS aperture address is truncated to addr[31:0]).
__device__ __forceinline__ u32 lds_byte_off(const void *p) {
  return (u32)(u64)p;
}

__global__ __launch_bounds__(POSES_PER_BLOCK)
void P3dPose_fk_kernel(const float *__restrict__ gChanges,
                       const float *__restrict__ gPrevLoc,
                       const float *__restrict__ gPrevRot,
                       float *__restrict__ out, long long nPose) {
  __shared__ float sC[POSES_PER_BLOCK * CH_F];   // changes -> rel_rot
  __shared__ float sL[POSES_PER_BLOCK * LOC_F];  // prev_loc -> abs_loc
  __shared__ float sR[POSES_PER_BLOCK * CH_F];   // prev_rot -> abs_rot

  const int lane = threadIdx.x;
  const long long pose0 = (long long)blockIdx.x * POSES_PER_BLOCK;

  const u32 offC = lds_byte_off(sC);
  const u32 offL = lds_byte_off(sL);
  const u32 offR = lds_byte_off(sR);

  // ---- TDM: bulk Global -> LDS for this block's 32-pose window -------------
  tdm_load(offC, gChanges + pose0 * CH_F,  POSES_PER_BLOCK * CH_F);
  tdm_load(offL, gPrevLoc + pose0 * LOC_F, POSES_PER_BLOCK * LOC_F);
  tdm_load(offR, gPrevRot + pose0 * CH_F,  POSES_PER_BLOCK * CH_F);
  __builtin_amdgcn_s_wait_tensorcnt(0);
  asm volatile("" ::: "memory");

  // ---- per-lane FK chain, in place in LDS ----------------------------------
  float *myC = sC + lane * CH_F;   // per-pose 234 floats
  float *myL = sL + lane * LOC_F;  // per-pose 78 floats
  float *myR = sR + lane * CH_F;

  static const int par[NB] = {-1, 0, 1, 2, 3, 4, 3, 6, 7, 8, 3, 10, 11, 12,
                              1, 14, 15, 16, 1, 18, 19, 20, 5, 5, 9, 13};

#pragma unroll
  for (int b = 0; b < NB; ++b) {
    float ch[9], pr[9], rel[9];
#pragma unroll
    for (int i = 0; i < 9; ++i) {
      ch[i] = myC[b * 9 + i];
      pr[i] = myR[b * 9 + i];
    }
    // rel_rot = changes @ prev_rot
#pragma unroll
    for (int i = 0; i < 3; ++i)
#pragma unroll
      for (int k = 0; k < 3; ++k)
        rel[i * 3 + k] = ch[i * 3 + 0] * pr[0 + k] +
                         ch[i * 3 + 1] * pr[3 + k] +
                         ch[i * 3 + 2] * pr[6 + k];
#pragma unroll
    for (int i = 0; i < 9; ++i) myC[b * 9 + i] = rel[i];  // slot C[b] := rel_rot

    const int p = par[b];
    if (p < 0) {
      // root: abs_rot = rel_rot; abs_loc = prev_loc (already in L[b])
#pragma unroll
      for (int i = 0; i < 9; ++i) myR[b * 9 + i] = rel[i];
    } else {
      float ap[9], alp[3], pl[3];
#pragma unroll
      for (int i = 0; i < 9; ++i) ap[i] = myR[p * 9 + i];  // abs_rot[parent]
#pragma unroll
      for (int k = 0; k < 3; ++k) {
        alp[k] = myL[p * 3 + k];                           // abs_loc[parent]
        pl[k]  = myL[b * 3 + k];                           // prev_loc[b]
      }
      // abs_rot[b] = rel_rot @ abs_rot[p]
#pragma unroll
      for (int i = 0; i < 3; ++i)
#pragma unroll
        for (int k = 0; k < 3; ++k)
          myR[b * 9 + i * 3 + k] = rel[i * 3 + 0] * ap[0 + k] +
                                   rel[i * 3 + 1] * ap[3 + k] +
                                   rel[i * 3 + 2] * ap[6 + k];
      // abs_loc[b] = prev_loc[b] @ abs_rot[p] + abs_loc[p]
#pragma unroll
      for (int k = 0; k < 3; ++k)
        myL[b * 3 + k] = pl[0] * ap[0 + k] + pl[1] * ap[3 + k] +
                         pl[2] * ap[6 + k] + alp[k];
    }
  }

  // Make this wave's DS stores visible before the TDM reads LDS.
  asm volatile("s_wait_dscnt 0x0" ::: "memory");

  // ---- TDM: bulk LDS -> Global (abs_loc, abs_rot, rel_rot in return order) -
  float *outLoc = out;                               // N * 78
  float *outRot = out + nPose * (long long)LOC_F;    // N * 234
  float *outRel = outRot + nPose * (long long)CH_F;  // N * 234

  tdm_store(offL, outLoc + pose0 * LOC_F, POSES_PER_BLOCK * LOC_F);
  tdm_store(offR, outRot + pose0 * CH_F,  POSES_PER_BLOCK * CH_F);
  tdm_store(offC, outRel + pose0 * CH_F,  POSES_PER_BLOCK * CH_F);
  // S_ENDPGM performs an implicit wait-idle (covers outstanding TENSORcnt).
}

extern "C" void kernel_launch(void* const* d_in, const int* in_sizes, int n_in,
                              void* d_out, int out_size, void* d_ws, size_t ws_size,
                              hipStream_t stream) {
  const float* gChanges = (const float*)d_in[0];  // (N, 26, 3, 3) f32
  const float* gPrevLoc = (const float*)d_in[1];  // (N, 26, 3)    f32
  const float* gPrevRot = (const float*)d_in[2];  // (N, 26, 3, 3) f32
  float* out = (float*)d_out;                     // abs_loc | abs_rot | rel_rot

  const long long nPose = (long long)in_sizes[1] / LOC_F;  // 131072
  const int blocks = (int)(nPose / POSES_PER_BLOCK);       // 4096

  P3dPose_fk_kernel<<<blocks, POSES_PER_BLOCK, 0, stream>>>(
      gChanges, gPrevLoc, gPrevRot, out, nPose);
}